// CTANet_42734924595559
// MI455X (gfx1250) — compile-verified
//
#include <hip/hip_runtime.h>

// ---------------------------------------------------------------------------
// Problem constants (from reference): B=64, C=256, HS=28 (P=784), KSEG=256,
// UP=16 -> query_mask is 64 x 448 x 448. Outputs: final mask (f32) then cos.
// ---------------------------------------------------------------------------
constexpr int B    = 64;
constexpr int C    = 256;
constexpr int P    = 28 * 28;      // 784 pixels
constexpr int KSEG = 256;
constexpr int QPIX = 448 * 448;    // 200704 per batch
constexpr int COS_OFF = B * QPIX;  // cos starts after final mask in d_out

// Workspace layout (in floats)
constexpr int OFF_LSUM   = 0;                     // 1 float (global label sum)
constexpr int OFF_SPROTO = 64;                    // B*C floats
constexpr int OFF_COUNTS = OFF_SPROTO + B * C;    // B*KSEG ints
constexpr int OFF_KEEPF  = OFF_COUNTS + B * KSEG; // B*KSEG floats
constexpr int OFF_QPROTO = OFF_KEEPF + B * KSEG;  // B*KSEG*C floats (16 MB)

typedef float v2f __attribute__((ext_vector_type(2)));
typedef float v8f __attribute__((ext_vector_type(8)));

// ---------------------------------------------------------------------------
// Kernel 1: global sum of small_s_label (B*P = 50176 elements), one block.
// ---------------------------------------------------------------------------
__global__ void lsum_kernel(const float* __restrict__ label, float* __restrict__ ws) {
    __shared__ float red[1024];
    int tid = threadIdx.x;
    float s = 0.f;
    for (int i = tid; i < B * P; i += 1024) s += label[i];
    red[tid] = s;
    __syncthreads();
    for (int off = 512; off > 0; off >>= 1) {
        if (tid < off) red[tid] += red[tid + off];
        __syncthreads();
    }
    if (tid == 0) ws[OFF_LSUM] = red[0];
}

// ---------------------------------------------------------------------------
// Kernel 2: s_proto[b,c] = sum_p s_feature[b,c,p]*label[b,p] / label_sum
// grid (32, B), 256 threads = 8 waves, one wave per channel (coalesced p).
// ---------------------------------------------------------------------------
__global__ void sproto_kernel(const float* __restrict__ sfeat,
                              const float* __restrict__ label,
                              float* __restrict__ ws) {
    __shared__ float lab[P];
    __shared__ float part[256];
    const int b   = blockIdx.y;
    const int cg  = blockIdx.x;          // channel group of 8
    const int tid = threadIdx.x;
    for (int i = tid; i < P; i += 256) lab[i] = label[b * P + i];
    __syncthreads();

    const int wave = tid >> 5;
    const int lane = tid & 31;
    const int c    = cg * 8 + wave;
    const float* row = sfeat + ((size_t)(b * C + c)) * P;
    float s = 0.f;
    for (int p = lane; p < P; p += 32) s += row[p] * lab[p];
    part[tid] = s;
    __syncthreads();

    if (tid < 8) {
        float t = 0.f;
        for (int l = 0; l < 32; ++l) t += part[tid * 32 + l];
        const float ls = ws[OFF_LSUM];
        ws[OFF_SPROTO + b * C + cg * 8 + tid] = t / ls;
    }
}

// ---------------------------------------------------------------------------
// Kernel 3: q_proto[b,k,c] = segment-mean of q_feature.
// grid (2, B), 128 threads. Each thread owns channel c = half*128 + tid and
// scatters its column into a 256x128 LDS accumulator -> no atomics needed
// (distinct LDS address per thread), bank-conflict-free (consecutive tid).
// Also computes integer segment counts (LDS int atomics, deterministic).
// ---------------------------------------------------------------------------
__global__ void qproto_kernel(const float* __restrict__ qfeat,
                              const int* __restrict__ segmask,
                              float* __restrict__ ws) {
    __shared__ float acc[KSEG * 128];   // 128 KB
    __shared__ int   seg[P];
    __shared__ int   cnt[KSEG];
    const int b    = blockIdx.y;
    const int half = blockIdx.x;
    const int tid  = threadIdx.x;

    for (int i = tid; i < P; i += 128) seg[i] = segmask[b * P + i];
    for (int i = tid; i < KSEG; i += 128) cnt[i] = 0;
    for (int i = tid; i < KSEG * 128; i += 128) acc[i] = 0.f;
    __syncthreads();

    for (int p = tid; p < P; p += 128) atomicAdd(&cnt[seg[p]], 1);

    const int c = half * 128 + tid;
    const float* row = qfeat + ((size_t)(b * C + c)) * P;
    for (int p = 0; p < P; ++p) {
        acc[seg[p] * 128 + tid] += row[p];
    }
    __syncthreads();

    float* qp = ws + OFF_QPROTO + (size_t)b * KSEG * C;
    for (int k = 0; k < KSEG; ++k) {
        const float d = (float)max(cnt[k], 1);
        qp[k * C + c] = acc[k * 128 + tid] / d;   // coalesced across tid
    }
    if (half == 0) {
        int* cw = (int*)(ws + OFF_COUNTS);
        for (int i = tid; i < KSEG; i += 128) cw[b * KSEG + i] = cnt[i];
    }
}

// ---------------------------------------------------------------------------
// Kernel 4: cos + keep logic. One block per batch, 256 threads (8 waves).
// num[b,k] = <s_proto[b], q_proto[b,k,:]> via V_WMMA_F32_16X16X4_F32:
//   A tile = 16 (k rows) x 4 (channels), B = s_proto broadcast over all 16 N.
//   ISA layout: A lane(0-15)=M rows, VGPR0/1 = K0/K1 (lanes 16-31 -> K2/K3);
//   B VGPR0 lanes0-15 = K0 (all N), lanes16-31 = K2; VGPR1 = K1/K3.
//   D VGPR r: lanes0-15 = row r, lanes16-31 = row r+8.
// Each wave handles its two k-tiles (wave, wave+8) with two independent
// accumulator chains sharing one B operand; unroll capped to 4 so the B
// operands come from ds_load_b64 each iteration instead of spilling to
// scratch (round-1 disasm showed scratch_load_b128 reloads).
// ---------------------------------------------------------------------------
__global__ void coskeep_kernel(float* __restrict__ ws, float* __restrict__ out) {
    __shared__ float sp[C];
    __shared__ float red[256];
    __shared__ float numArr[KSEG];
    __shared__ float qn[KSEG];
    __shared__ int   ired[256];
    __shared__ float snormSh;

    const int b   = blockIdx.x;
    const int tid = threadIdx.x;

    // s_proto into LDS + its norm (tree reduce)
    const float sv = ws[OFF_SPROTO + b * C + tid];
    sp[tid]  = sv;
    red[tid] = sv * sv;
    __syncthreads();
    for (int off = 128; off > 0; off >>= 1) {
        if (tid < off) red[tid] += red[tid + off];
        __syncthreads();
    }
    if (tid == 0) snormSh = sqrtf(red[0]);
    __syncthreads();

    // WMMA matvec: wave w computes k-tiles w and w+8 with dual acc chains.
    const int wave = tid >> 5;
    const int lane = tid & 31;
    const int m    = lane & 15;
    const int hi   = lane >> 4;      // 0: K pair {0,1}, 1: K pair {2,3}
    const float* qp = ws + OFF_QPROTO + (size_t)b * KSEG * C;

    {
        v8f acc0 = {};
        v8f acc1 = {};
        const float* row0 = qp + (wave * 16 + m) * C + 2 * hi;
        const float* row1 = qp + ((wave + 8) * 16 + m) * C + 2 * hi;
        const float* bptr = sp + 2 * hi;
#pragma unroll 4
        for (int cb = 0; cb < C; cb += 4) {
            const v2f a0 = *(const v2f*)(row0 + cb);   // global_load_b64
            const v2f a1 = *(const v2f*)(row1 + cb);   // global_load_b64
            const v2f bb = *(const v2f*)(bptr + cb);   // ds_load_b64
            acc0 = __builtin_amdgcn_wmma_f32_16x16x4_f32(
                false, a0, false, bb, (short)0, acc0, false, false);
            acc1 = __builtin_amdgcn_wmma_f32_16x16x4_f32(
                false, a1, false, bb, (short)0, acc1, false, false);
        }
        if (m == 0) {  // column N=0 lives in lanes 0 (rows 0-7) and 16 (rows 8-15)
            for (int r = 0; r < 8; ++r) {
                numArr[wave * 16 + hi * 8 + r]       = acc0[r];
                numArr[(wave + 8) * 16 + hi * 8 + r] = acc1[r];
            }
        }
    }

    // per-k q_proto row norm
    {
        const float* row = qp + tid * C;
        float s = 0.f;
        for (int c = 0; c < C; ++c) s += row[c] * row[c];
        qn[tid] = sqrtf(s);
    }
    __syncthreads();

    const int* cw = (const int*)(ws + OFF_COUNTS);
    const int cnt = cw[b * KSEG + tid];
    const float den  = fmaxf(snormSh * qn[tid], 1e-8f);
    const float cosv = numArr[tid] / den;
    out[COS_OFF + b * KSEG + tid] = cosv;

    const int present = (cnt > 0 && tid > 0) ? 1 : 0;
    ired[tid] = present;
    __syncthreads();
    for (int off = 128; off > 0; off >>= 1) {
        if (tid < off) ired[tid] += ired[tid + off];
        __syncthreads();
    }
    const int npres = ired[0];
    const int keep  = present && (cosv >= 0.f || tid == 255 || npres == 1);
    ws[OFF_KEEPF + b * KSEG + tid] = keep ? 255.0f : 0.0f;
}

// ---------------------------------------------------------------------------
// Kernel 5: the big gather (dominant traffic). 256-float LUT in LDS,
// int4 -> float4 vectorized. grid (196, B), 256 threads, 4 px/thread.
// ---------------------------------------------------------------------------
__global__ void gather_kernel(const int* __restrict__ qmask,
                              const float* __restrict__ ws,
                              float* __restrict__ out) {
    __shared__ float kf[KSEG];
    const int b   = blockIdx.y;
    const int tid = threadIdx.x;
    kf[tid] = ws[OFF_KEEPF + b * KSEG + tid];
    __syncthreads();

    const size_t base = (size_t)b * QPIX + ((size_t)blockIdx.x * 256 + tid) * 4;
    const int4 v = *(const int4*)(qmask + base);
    float4 o;
    o.x = kf[v.x];
    o.y = kf[v.y];
    o.z = kf[v.z];
    o.w = kf[v.w];
    *(float4*)(out + base) = o;
}

// ---------------------------------------------------------------------------
extern "C" void kernel_launch(void* const* d_in, const int* in_sizes, int n_in,
                              void* d_out, int out_size, void* d_ws, size_t ws_size,
                              hipStream_t stream) {
    const float* s_feature  = (const float*)d_in[0];
    const float* s_label    = (const float*)d_in[1];
    const float* q_feature  = (const float*)d_in[2];
    const int*   small_qm   = (const int*)d_in[3];
    const int*   query_mask = (const int*)d_in[4];
    float* out = (float*)d_out;
    float* ws  = (float*)d_ws;

    lsum_kernel  <<<1, 1024, 0, stream>>>(s_label, ws);
    sproto_kernel<<<dim3(C / 8, B), 256, 0, stream>>>(s_feature, s_label, ws);
    qproto_kernel<<<dim3(2, B), 128, 0, stream>>>(q_feature, small_qm, ws);
    coskeep_kernel<<<B, 256, 0, stream>>>(ws, out);
    gather_kernel<<<dim3(QPIX / 1024, B), 256, 0, stream>>>(query_mask, ws, out);
}